// MaxUnpooling2D_19292993093787
// MI455X (gfx1250) — compile-verified
//
#include <hip/hip_runtime.h>

// MaxUnpooling2D, fused gather formulation for MI455X (gfx1250, wave32).
//
// updates [16,64,64,256] f32, mask [16,64,64,256] i32 (flat index into
// out [16,128,128,256] f32). Each input cell (b,h,w,c) can only scatter into
// the 2x2 window rooted at (2h,2w) of channel c, so each thread OWNS the 4
// window positions for 4 consecutive channels: load updates+mask once, write
// all 4 positions (value where mask matches, 0 elsewhere). Output written
// exactly once, fully coalesced b128 stores, no atomics, no zero pass.
//
// Roofline: 0 useful FLOPs/byte -> pure HBM bandwidth problem. Traffic =
// 128 MiB read + 256 MiB write = 384 MiB -> ~16.5 us floor @ 23.3 TB/s.
// Output (256 MiB) exceeds the 192 MB L2 -> non-temporal (th:NT) stores;
// inputs stream once -> non-temporal b128 loads + WGP-scope prefetch ahead.

typedef __attribute__((ext_vector_type(4))) float v4f;
typedef __attribute__((ext_vector_type(4))) int   v4i;

namespace {
constexpr int B  = 16;
constexpr int H  = 64;
constexpr int W  = 64;
constexpr int C  = 256;
constexpr int W2 = 128;                   // W * 2
constexpr int ROW = W2 * C;               // 32768 floats per output row
constexpr int NVEC = B * H * W * C / 4;   // 4,194,304 vec4 work items
constexpr int PREFETCH_VEC = 16384;       // ~256 KiB ahead per stream
}

__global__ __launch_bounds__(256) void maxunpool2d_fused_kernel(
    const float* __restrict__ upd,
    const int*   __restrict__ mask,
    float*       __restrict__ out)
{
    // Grid exactly covers NVEC -> no bounds check needed.
    const int j = blockIdx.x * blockDim.x + threadIdx.x;  // vec4 id

    // Decode (b,h,w,q): 64 vec4 groups per channel dim, then w,h,b.
    const int q    = j & 63;          // vec4 group within C
    const int cell = j >> 6;          // (b,h,w) cell id
    const int w    = cell & 63;
    const int h    = (cell >> 6) & 63;
    const int b    = cell >> 12;

    // Branch-free L2 warm-up ahead of the streaming front
    // (locality 3 -> WGP scope: fills all cache levels).
    const int jp = min(j + PREFETCH_VEC, NVEC - 1);
    __builtin_prefetch(upd  + (size_t)jp * 4, 0, 3);
    __builtin_prefetch(mask + (size_t)jp * 4, 0, 3);

    const v4f u = __builtin_nontemporal_load((const v4f*)upd  + j);
    const v4i m = __builtin_nontemporal_load((const v4i*)mask + j);

    // Flat output offset of window position (2h, 2w), channels [4q, 4q+4).
    const int p00 = ((b * (2 * H) + 2 * h) * W2 + 2 * w) * C + q * 4;

    v4f z00, z01, z10, z11;
#pragma unroll
    for (int k = 0; k < 4; ++k) {
        const int   cand = p00 + k;
        const float v    = u[k];
        const int   mk   = m[k];
        z00[k] = (mk == cand           ) ? v : 0.0f;  // (2h,   2w  )
        z01[k] = (mk == cand + C       ) ? v : 0.0f;  // (2h,   2w+1)
        z10[k] = (mk == cand + ROW     ) ? v : 0.0f;  // (2h+1, 2w  )
        z11[k] = (mk == cand + ROW + C ) ? v : 0.0f;  // (2h+1, 2w+1)
    }

    // Four contiguous-per-wave b128 streaming stores: each wave writes
    // 4 x 512 B fully-populated line segments; output covered exactly once.
    __builtin_nontemporal_store(z00, (v4f*)(out + p00));
    __builtin_nontemporal_store(z01, (v4f*)(out + p00 + C));
    __builtin_nontemporal_store(z10, (v4f*)(out + p00 + ROW));
    __builtin_nontemporal_store(z11, (v4f*)(out + p00 + ROW + C));
}

extern "C" void kernel_launch(void* const* d_in, const int* in_sizes, int n_in,
                              void* d_out, int out_size, void* d_ws, size_t ws_size,
                              hipStream_t stream) {
    (void)in_sizes; (void)n_in; (void)out_size; (void)d_ws; (void)ws_size;
    const float* upd  = (const float*)d_in[0];
    const int*   mask = (const int*)d_in[1];
    float*       out  = (float*)d_out;

    constexpr int threads = 256;             // 8 wave32 waves per workgroup
    constexpr int blocks  = NVEC / threads;  // 16384, exact cover
    maxunpool2d_fused_kernel<<<blocks, threads, 0, stream>>>(upd, mask, out);
}